// SparseLoss_78056735637528
// MI455X (gfx1250) — compile-verified
//
#include <hip/hip_runtime.h>

// ---- types ------------------------------------------------------------
typedef __attribute__((ext_vector_type(4))) float f32x4;
typedef __attribute__((ext_vector_type(4))) int   i32x4;
typedef __attribute__((ext_vector_type(2))) float v2f;
typedef __attribute__((ext_vector_type(8))) float v8f;

#define EPSF 1e-10f
#define P1_BLOCKS  2048
#define P1_THREADS 256

// ---------------------------------------------------------------------
// Pass 1: streaming partial sums. HBM-bound: 2x B128 NT loads per thread
// iteration, 1 log per element, branchless select, grid-stride.
// Writes one deterministic partial per block into d_ws.
// ---------------------------------------------------------------------
__global__ __launch_bounds__(P1_THREADS)
void sparse_loss_pass1(const float* __restrict__ pred,
                       const int*   __restrict__ y,
                       const int*   __restrict__ lamda_p,
                       float*       __restrict__ partials,
                       int n4, int n)
{
    const float lam = (float)lamda_p[0];
    const f32x4* __restrict__ p4 = (const f32x4*)pred;
    const i32x4* __restrict__ y4 = (const i32x4*)y;

    float acc = 0.0f;
    const int stride = gridDim.x * blockDim.x;
    for (int i = blockIdx.x * blockDim.x + threadIdx.x; i < n4; i += stride) {
        // 16B non-temporal loads: global_load_b128 th:TH_LOAD_NT (stream once,
        // keep L2 clean -- no reuse across the 327MB working set)
        f32x4 p  = __builtin_nontemporal_load(&p4[i]);
        i32x4 yy = __builtin_nontemporal_load(&y4[i]);
#pragma unroll
        for (int j = 0; j < 4; ++j) {
            float pv  = p[j];
            bool  pos = (yy[j] != 0);
            float x   = pos ? (pv + EPSF) : ((1.0f - pv) + EPSF);
            float w   = pos ? lam : 1.0f;
            acc += w * __logf(x);     // elem = -w*log(x); negate once at the end
        }
    }
    // scalar tail (n % 4), handled by one thread; n here is divisible by 4
    // but stay correct for any n.
    if (blockIdx.x == 0 && threadIdx.x == 0) {
        for (int i = n4 * 4; i < n; ++i) {
            float pv  = pred[i];
            bool  pos = (y[i] != 0);
            float x   = pos ? (pv + EPSF) : ((1.0f - pv) + EPSF);
            float w   = pos ? lam : 1.0f;
            acc += w * __logf(x);
        }
    }

    // wave32 xor-shuffle reduction (warpSize == 32 on gfx1250)
    acc += __shfl_xor(acc, 16, 32);
    acc += __shfl_xor(acc,  8, 32);
    acc += __shfl_xor(acc,  4, 32);
    acc += __shfl_xor(acc,  2, 32);
    acc += __shfl_xor(acc,  1, 32);

    __shared__ float red[P1_THREADS / 32];
    const int lane = threadIdx.x & 31;
    const int wid  = threadIdx.x >> 5;
    if (lane == 0) red[wid] = acc;
    __syncthreads();
    if (threadIdx.x == 0) {
        float s = 0.0f;
#pragma unroll
        for (int w = 0; w < P1_THREADS / 32; ++w) s += red[w];
        partials[blockIdx.x] = s;   // deterministic per-block partial
    }
}

// ---------------------------------------------------------------------
// Pass 2: single wave32. Each lane accumulates nparts/32 partials, then a
// single V_WMMA_F32_16X16X4_F32 with A = ones(16x4) performs the cross-lane
// sum: D[m][n] = sum_k B[k][n]. The 2-VGPR B operand maps bijectively onto
// {32 lanes} x {2 slots}; with B.vgpr0 = s and B.vgpr1 = 0 every lane's
// partial lands in exactly one (k,n) slot, so Sum_n D[0][n] = Sum_lanes s.
// D row 0 lives in VGPR0 of lanes 0..15 (row 8, identical values, in lanes
// 16..31), so 4 shfl_xor steps finish the reduction.
// ---------------------------------------------------------------------
__global__ __launch_bounds__(32)
void sparse_loss_pass2(const float* __restrict__ partials,
                       int nparts,
                       float* __restrict__ out,
                       float inv_n)
{
    const int lane = threadIdx.x;   // exactly one wave32, EXEC all 1s (WMMA req)
    float s = 0.0f;
    for (int i = lane; i < nparts; i += 32) s += partials[i];  // uniform trips

    v2f a; a[0] = 1.0f; a[1] = 1.0f;   // A = ones(16x4)
    v2f b; b[0] = s;    b[1] = 0.0f;   // lane partials occupy 32 of 64 B slots
    v8f c = {};
    // (neg_a, A, neg_b, B, c_mod, C, reuse_a, reuse_b)
    v8f d = __builtin_amdgcn_wmma_f32_16x16x4_f32(
        false, a, false, b, (short)0, c, false, false);

    float col = d[0];                  // column sums, function of (lane % 16)
    col += __shfl_xor(col, 8, 32);
    col += __shfl_xor(col, 4, 32);
    col += __shfl_xor(col, 2, 32);
    col += __shfl_xor(col, 1, 32);     // every lane now holds Sum_n D[0][n]

    if (lane == 0) out[0] = -col * inv_n;   // mean with the deferred negation
}

// ---------------------------------------------------------------------
extern "C" void kernel_launch(void* const* d_in, const int* in_sizes, int n_in,
                              void* d_out, int out_size, void* d_ws, size_t ws_size,
                              hipStream_t stream)
{
    const float* pred  = (const float*)d_in[0];
    const int*   y     = (const int*)  d_in[1];
    const int*   lamda = (const int*)  d_in[2];
    float*       out   = (float*)d_out;
    float*       parts = (float*)d_ws;       // P1_BLOCKS floats = 8 KB scratch

    const int n  = in_sizes[0];              // 4096 * 10000
    const int n4 = n >> 2;

    sparse_loss_pass1<<<P1_BLOCKS, P1_THREADS, 0, stream>>>(
        pred, y, lamda, parts, n4, n);
    sparse_loss_pass2<<<1, 32, 0, stream>>>(
        parts, P1_BLOCKS, out, 1.0f / (float)n);
}